// CBAMSpaceMask_20109036880328
// MI455X (gfx1250) — compile-verified
//
#include <hip/hip_runtime.h>
#include <cstdint>

// CBAM spatial mask, fused: 3x3 max/avg pool -> 6ch->1ch 3x3 conv -> LeakyReLU
// -> sigmoid -> broadcast to 3 channels. Memory-bound (~201 MB @ 23.3 TB/s).
// Input tile staged in LDS via gfx1250 async global->LDS loads.

#define NCH     3
#define TSTEPS  16
#define HDIM    256
#define WDIM    256
#define TILE    32
#define IN_W    (TILE + 4)   // 36: input tile with halo 2 (pool+conv)
#define PL_W    (TILE + 2)   // 34: pooled tile with halo 1 (conv)
#define NTHREADS 256

typedef __attribute__((address_space(1))) int g_int;
typedef __attribute__((address_space(3))) int l_int;

__device__ __forceinline__ void async_load_f32(const float* g, float* lds) {
  unsigned lds_off = (unsigned)(unsigned long long)(uintptr_t)lds;  // flat[31:0] == LDS byte addr
#if __has_builtin(__builtin_amdgcn_global_load_async_to_lds_b32)
  __builtin_amdgcn_global_load_async_to_lds_b32(
      (g_int*)(uintptr_t)g,
      (l_int*)(uintptr_t)lds_off,
      0, 0);
#else
  asm volatile("global_load_async_to_lds_b32 %0, %1, off"
               :: "v"(lds_off), "v"((unsigned long long)(uintptr_t)g)
               : "memory");
#endif
}

__device__ __forceinline__ void wait_async0() {
#if __has_builtin(__builtin_amdgcn_s_wait_asynccnt)
  __builtin_amdgcn_s_wait_asynccnt(0);
#else
  asm volatile("s_wait_asynccnt 0" ::: "memory");
#endif
}

__global__ __launch_bounds__(NTHREADS)
void cbam_space_mask_kernel(const float* __restrict__ x,
                            const float* __restrict__ cw,
                            const float* __restrict__ cb,
                            float* __restrict__ out) {
  __shared__ float s_in[NCH][IN_W][IN_W];   // 15552 B
  __shared__ float s_mx[NCH][PL_W][PL_W];   // 13872 B
  __shared__ float s_av[NCH][PL_W][PL_W];   // 13872 B
  __shared__ float s_w[56];                 // 54 weights + bias

  const int tid = threadIdx.x;
  const int tilesX = WDIM / TILE;                    // 8
  const int tilesY = HDIM / TILE;                    // 8
  int bx = blockIdx.x;
  int tw = bx % tilesX;
  int th = (bx / tilesX) % tilesY;
  int bt = bx / (tilesX * tilesY);                   // b*TSTEPS + t
  const int h0 = th * TILE, w0 = tw * TILE;

  const long plane = (long)HDIM * WDIM;
  const float* inb  = x   + (long)bt * NCH * plane;
  float*       outb = out + (long)bt * NCH * plane;

  // stage conv weights + bias in LDS
  if (tid < 54) s_w[tid] = cw[tid];
  if (tid == 0) s_w[54] = cb[0];

  // --- Phase 1: async global -> LDS tile fill (36x36 x 3 channels) ---
  // Border addresses are clamped; clamped garbage is never read because the
  // pool phase bounds-checks global coordinates before touching s_in.
  for (int e = tid; e < NCH * IN_W * IN_W; e += NTHREADS) {
    int c = e / (IN_W * IN_W);
    int r = (e / IN_W) % IN_W;
    int j = e % IN_W;
    int gh = h0 - 2 + r;
    int gw = w0 - 2 + j;
    int ghc = gh < 0 ? 0 : (gh > HDIM - 1 ? HDIM - 1 : gh);
    int gwc = gw < 0 ? 0 : (gw > WDIM - 1 ? WDIM - 1 : gw);
    async_load_f32(inb + (long)c * plane + (long)ghc * WDIM + gwc, &s_in[c][r][j]);
  }
  wait_async0();        // each wave drains its own ASYNCcnt
  __syncthreads();      // then all LDS writes are visible workgroup-wide

  // --- Phase 2: 3x3 max/avg pool into 34x34 pooled tiles ---
  const bool interior = (h0 >= 2) && (h0 + TILE + 2 <= HDIM) &&
                        (w0 >= 2) && (w0 + TILE + 2 <= WDIM);
  for (int q = tid; q < NCH * PL_W * PL_W; q += NTHREADS) {
    int c  = q / (PL_W * PL_W);
    int rr = q - c * (PL_W * PL_W);
    int pr = rr / PL_W;
    int pc = rr - pr * PL_W;
    float mx, av;
    if (interior) {
      float sum = 0.0f;
      mx = -__builtin_inff();
#pragma unroll
      for (int i = 0; i < 3; ++i)
#pragma unroll
        for (int j = 0; j < 3; ++j) {
          float v = s_in[c][pr + i][pc + j];
          sum += v;
          mx = fmaxf(mx, v);
        }
      av = sum * (1.0f / 9.0f);
    } else {
      int ph = h0 - 1 + pr, pw = w0 - 1 + pc;
      if ((unsigned)ph >= (unsigned)HDIM || (unsigned)pw >= (unsigned)WDIM) {
        mx = 0.0f; av = 0.0f;            // conv zero-padding on pooled map
      } else {
        float sum = 0.0f;
        mx = -__builtin_inff();          // pool pad: -inf for max, 0 for sum
#pragma unroll
        for (int i = 0; i < 3; ++i) {
          int ih = ph - 1 + i;
#pragma unroll
          for (int j = 0; j < 3; ++j) {
            int iw = pw - 1 + j;
            if ((unsigned)ih < (unsigned)HDIM && (unsigned)iw < (unsigned)WDIM) {
              float v = s_in[c][pr + i][pc + j];
              sum += v;
              mx = fmaxf(mx, v);
            }
          }
        }
        av = sum * (1.0f / 9.0f);        // count_include_pad=True -> /9 always
      }
    }
    s_mx[c][pr][pc] = mx;
    s_av[c][pr][pc] = av;
  }
  __syncthreads();

  // --- Phase 3: 6ch 3x3 conv + bias + LeakyReLU + sigmoid, broadcast x3 ---
  for (int px = tid; px < TILE * TILE; px += NTHREADS) {
    int oh = px >> 5;
    int ow = px & (TILE - 1);
    float acc = s_w[54];
#pragma unroll
    for (int c = 0; c < NCH; ++c)
#pragma unroll
      for (int ky = 0; ky < 3; ++ky)
#pragma unroll
        for (int kx = 0; kx < 3; ++kx) {
          // interleave: conv input channel 2c = max(c), 2c+1 = avg(c)
          acc = fmaf(s_w[((2 * c    ) * 3 + ky) * 3 + kx], s_mx[c][oh + ky][ow + kx], acc);
          acc = fmaf(s_w[((2 * c + 1) * 3 + ky) * 3 + kx], s_av[c][oh + ky][ow + kx], acc);
        }
    float y  = acc >= 0.0f ? acc : 0.01f * acc;       // LeakyReLU
    float sg = 1.0f / (1.0f + __expf(-y));            // sigmoid (v_exp_f32)
    long off = (long)(h0 + oh) * WDIM + (w0 + ow);
    outb[off]             = sg;                        // lane-coalesced rows
    outb[plane + off]     = sg;
    outb[2 * plane + off] = sg;
  }
}

extern "C" void kernel_launch(void* const* d_in, const int* in_sizes, int n_in,
                              void* d_out, int out_size, void* d_ws, size_t ws_size,
                              hipStream_t stream) {
  (void)n_in; (void)d_ws; (void)ws_size; (void)out_size;
  const float* x  = (const float*)d_in[0];
  const float* cw = (const float*)d_in[1];
  const float* cb = (const float*)d_in[2];
  float* out = (float*)d_out;

  int B = in_sizes[0] / (TSTEPS * NCH * HDIM * WDIM);   // 8 for the reference
  int grid = B * TSTEPS * (HDIM / TILE) * (WDIM / TILE);
  hipLaunchKernelGGL(cbam_space_mask_kernel, dim3(grid), dim3(NTHREADS), 0, stream,
                     x, cw, cb, out);
}